// DigitCapsuleLayer_15410342658245
// MI455X (gfx1250) — compile-verified
//
#include <hip/hip_runtime.h>
#include <cstddef>

// ---------------------------------------------------------------------------
// Capsule dynamic routing for MI455X (gfx1250, wave32, WMMA).
//
// u_hat (B=128, R=2048, C=10, O=16) f32 = 167.8 MB -> fits global L2 (192MB).
// Strategy: materialize u_hat once via V_WMMA_F32_16X16X4_F32 (f32 is the
// right precision: problem is bandwidth bound, ~671 MFLOP total for the GEMM),
// then run 3 routing iterations against L2-resident u_hat.
// ---------------------------------------------------------------------------

#define BATCH      128
#define NUM_ROUTES 2048
#define NUM_CAPS   10
#define IN_DIM     8
#define OUT_DIM    16
#define ROUT_ITERS 3

typedef __attribute__((ext_vector_type(2))) float v2f;
typedef __attribute__((ext_vector_type(8))) float v8f;

// ---------------------------------------------------------------------------
// Kernel 1: u_hat[b,r,c,o] = sum_i W[r,c,o,i] * x[b,r,i]
// One block (8 wave32) per route r. Wave w handles batch rows [16w, 16w+16).
// Per-route GEMM: M=128 (batch), N=160 (c*16+o), K=8 (i) via 2x WMMA 16x16x4.
//
// A layout (16x4 f32, ISA 7.12.2): lane L -> row M=L%16; VGPR0 holds K=2*(L/16),
// VGPR1 holds K=2*(L/16)+1  => one contiguous float2 load of x per k-half.
// B layout (4x16 f32, mirrored):  lane L -> col N=L%16; VGPR0/1 hold the same
// K pair                         => one contiguous float2 load of W per k-half.
// D layout: VGPR g -> (M = g + 8*(L/16), N = L%16).
// ---------------------------------------------------------------------------
__global__ __launch_bounds__(256)
void uhat_wmma_kernel(const float* __restrict__ x,
                      const float* __restrict__ W,
                      float* __restrict__ u_hat)
{
    const int r    = blockIdx.x;
    const int wave = threadIdx.x >> 5;   // 0..7 -> m-tile
    const int lane = threadIdx.x & 31;
    const int lrow = lane & 15;          // M row (A/D) or N col (B/D) in tile
    const int lhi  = lane >> 4;          // k-pair selector

    const int m_base = wave * 16;

    // A fragments for both K-halves (reused across all 10 capsules).
    const float* xrow = x + ((size_t)(m_base + lrow) * NUM_ROUTES + r) * IN_DIM;
    const v2f a0 = *(const v2f*)(xrow + 2 * lhi);       // i in {0..3}
    const v2f a1 = *(const v2f*)(xrow + 4 + 2 * lhi);   // i in {4..7}

    for (int c = 0; c < NUM_CAPS; ++c) {
        const float* wrow =
            W + (((size_t)r * NUM_CAPS + c) * OUT_DIM + lrow) * IN_DIM;
        const v2f b0 = *(const v2f*)(wrow + 2 * lhi);
        const v2f b1 = *(const v2f*)(wrow + 4 + 2 * lhi);

        v8f acc = {};
        // 8-arg form: (neg_a, A, neg_b, B, c_mod, C, reuse_a, reuse_b)
        acc = __builtin_amdgcn_wmma_f32_16x16x4_f32(
                  false, a0, false, b0, (short)0, acc, false, false);
        acc = __builtin_amdgcn_wmma_f32_16x16x4_f32(
                  false, a1, false, b1, (short)0, acc, false, false);

        // Store D: lane halves write 64B-contiguous (o=0..15) runs per row b.
        #pragma unroll
        for (int g = 0; g < 8; ++g) {
            const int b = m_base + g + 8 * lhi;
            u_hat[(((size_t)b * NUM_ROUTES + r) * NUM_CAPS + c) * OUT_DIM + lrow]
                = acc[g];
        }
    }
}

// ---------------------------------------------------------------------------
// Kernel 2: zero routing logits b_ij (workspace is poisoned, must re-init
// every launch for determinism).
// ---------------------------------------------------------------------------
__global__ void zero_kernel(float* __restrict__ p, int n)
{
    const int i = blockIdx.x * blockDim.x + threadIdx.x;
    if (i < n) p[i] = 0.0f;
}

// ---------------------------------------------------------------------------
// Kernel 3: c_ij = softmax(b_ij) over the capsule axis (10 values / (b,r)).
// ---------------------------------------------------------------------------
__global__ void softmax_kernel(const float* __restrict__ b_ij,
                               float* __restrict__ c_ij, int n_br)
{
    const int i = blockIdx.x * blockDim.x + threadIdx.x;
    if (i >= n_br) return;
    const float* bp = b_ij + (size_t)i * NUM_CAPS;
    float m = bp[0];
    #pragma unroll
    for (int c = 1; c < NUM_CAPS; ++c) m = fmaxf(m, bp[c]);
    float e[NUM_CAPS];
    float s = 0.0f;
    #pragma unroll
    for (int c = 0; c < NUM_CAPS; ++c) { e[c] = __expf(bp[c] - m); s += e[c]; }
    const float inv = 1.0f / s;
    float* cp = c_ij + (size_t)i * NUM_CAPS;
    #pragma unroll
    for (int c = 0; c < NUM_CAPS; ++c) cp[c] = e[c] * inv;
}

// ---------------------------------------------------------------------------
// Kernel 4: s[b,c,o] = sum_r c_ij[b,r,c] * u_hat[b,r,c,o]; v = squash(s).
// One block per (b,c). Thread t: o = t&15, r-chunk = t>>4 (16 chunks).
// LDS tree reduction over chunks, then squash on lanes 0..15, write d_out.
// ---------------------------------------------------------------------------
__global__ __launch_bounds__(256)
void s_squash_kernel(const float* __restrict__ u_hat,
                     const float* __restrict__ c_ij,
                     float* __restrict__ v_out)
{
    __shared__ float red[256];
    const int bc = blockIdx.x;
    const int b  = bc / NUM_CAPS;
    const int c  = bc % NUM_CAPS;
    const int t     = threadIdx.x;
    const int o     = t & 15;
    const int chunk = t >> 4;

    float acc = 0.0f;
    for (int r = chunk; r < NUM_ROUTES; r += 16) {
        const size_t brc = ((size_t)b * NUM_ROUTES + r) * NUM_CAPS + c;
        acc += c_ij[brc] * u_hat[brc * OUT_DIM + o];
    }
    red[t] = acc;
    __syncthreads();
    #pragma unroll
    for (int s = 128; s >= 16; s >>= 1) {
        if (t < s) red[t] += red[t + s];
        __syncthreads();
    }
    if (t < 16) {
        const float sv = red[t];
        float sq = 0.0f;
        #pragma unroll
        for (int k = 0; k < 16; ++k) { const float q = red[k]; sq += q * q; }
        const float norm  = sqrtf(sq + 1e-8f);
        const float scale = sq / (1.0f + sq);
        v_out[((size_t)b * NUM_CAPS + c) * OUT_DIM + t] = scale * sv / norm;
    }
}

// ---------------------------------------------------------------------------
// Kernel 5: b_ij[b,r,c] += dot_o(u_hat[b,r,c,:], v[b,c,:])
// ---------------------------------------------------------------------------
__global__ void agree_kernel(const float* __restrict__ u_hat,
                             const float* __restrict__ v,
                             float* __restrict__ b_ij, int n_brc)
{
    const int i = blockIdx.x * blockDim.x + threadIdx.x;
    if (i >= n_brc) return;
    const int c = i % NUM_CAPS;
    const int b = i / (NUM_CAPS * NUM_ROUTES);
    const float* up = u_hat + (size_t)i * OUT_DIM;
    const float* vp = v + ((size_t)b * NUM_CAPS + c) * OUT_DIM;
    float dot = 0.0f;
    #pragma unroll
    for (int o = 0; o < OUT_DIM; ++o) dot += up[o] * vp[o];
    b_ij[i] += dot;
}

// ---------------------------------------------------------------------------
extern "C" void kernel_launch(void* const* d_in, const int* in_sizes, int n_in,
                              void* d_out, int out_size, void* d_ws, size_t ws_size,
                              hipStream_t stream)
{
    const float* x = (const float*)d_in[0];  // (B, R, IN)
    const float* W = (const float*)d_in[1];  // (R, C, O, IN)
    float* out = (float*)d_out;              // (B, C, O)

    // Workspace layout (~189 MB total):
    float* u_hat = (float*)d_ws;                                         // B*R*C*O
    float* b_ij  = u_hat + (size_t)BATCH * NUM_ROUTES * NUM_CAPS * OUT_DIM; // B*R*C
    float* c_ij  = b_ij + (size_t)BATCH * NUM_ROUTES * NUM_CAPS;            // B*R*C

    const int n_brc = BATCH * NUM_ROUTES * NUM_CAPS;   // 2,621,440
    const int n_br  = BATCH * NUM_ROUTES;              // 262,144

    // 1) u_hat via WMMA (one block per route).
    uhat_wmma_kernel<<<NUM_ROUTES, 256, 0, stream>>>(x, W, u_hat);

    // 2) b_ij = 0 (must re-init every call).
    zero_kernel<<<(n_brc + 255) / 256, 256, 0, stream>>>(b_ij, n_brc);

    // 3) dynamic routing, 3 iterations; v lives in d_out each iteration.
    for (int it = 0; it < ROUT_ITERS; ++it) {
        softmax_kernel<<<(n_br + 255) / 256, 256, 0, stream>>>(b_ij, c_ij, n_br);
        s_squash_kernel<<<BATCH * NUM_CAPS, 256, 0, stream>>>(u_hat, c_ij, out);
        if (it < ROUT_ITERS - 1) {
            agree_kernel<<<(n_brc + 255) / 256, 256, 0, stream>>>(u_hat, out,
                                                                  b_ij, n_brc);
        }
    }
}